// PointsNonMaxSuppression_63196148794003
// MI455X (gfx1250) — compile-verified
//
#include <hip/hip_runtime.h>
#include <cstdint>

// Problem shape (fixed by the reference): points (B=4, C=84, H=512, W=512) f32.
#define NB 4
#define NC 84
#define NCP 80          // channels participating in the prob max (C - 4)
#define NH 512
#define NW 512
#define NHW (NH * NW)   // 262144
#define NSTAGE 4        // async pipeline depth (16 KB LDS per block)

typedef float f4 __attribute__((ext_vector_type(4)));

// ---- gfx1250 async global->LDS via inline asm (portable across toolchains) ----
// Per-lane: copies 16 bytes from the global address g to the LDS byte address lds_byte; tracked by ASYNCcnt.
__device__ __forceinline__ void async_g2l_b128(const float* g, unsigned lds_byte) {
  asm volatile("global_load_async_to_lds_b128 %0, %1, off"
               :
               : "v"(lds_byte), "v"(g)
               : "memory");
}
__device__ __forceinline__ void wait_async_le3() {
  asm volatile("s_wait_asynccnt 0x3" ::: "memory");
}
__device__ __forceinline__ void wait_async_0() {
  asm volatile("s_wait_asynccnt 0x0" ::: "memory");
}
__device__ __forceinline__ void wait_ds_0() {
  asm volatile("s_wait_dscnt 0x0" ::: "memory");
}

// Kernel 1: probs[b,h,w] = max_{c<80} points[b,c,h,w]; 4 pixels / thread.
__global__ __launch_bounds__(256) void probs_max_kernel(
    const float* __restrict__ in, float* __restrict__ probs) {
  const int g   = blockIdx.x * 256 + threadIdx.x;  // 4-pixel group id
  const int p0  = g << 2;
  const int b   = p0 / NHW;
  const int rem = p0 - b * NHW;
  const float* base = in + (size_t)b * NC * NHW + rem;

  f4 m0 = *(const f4*)base;
  f4 m1 = *(const f4*)(base + (size_t)1 * NHW);
#pragma unroll 8
  for (int c = 2; c < NCP; c += 2) {
    f4 v0 = *(const f4*)(base + (size_t)c * NHW);
    f4 v1 = *(const f4*)(base + (size_t)(c + 1) * NHW);
    m0 = __builtin_elementwise_max(m0, v0);
    m1 = __builtin_elementwise_max(m1, v1);
  }
  *(f4*)(probs + p0) = __builtin_elementwise_max(m0, m1);
}

// Kernel 2: 3x3 NMS mask from probs, then out[b,c,h,w] = in * mask for all 84
// channels, streamed through a 4-stage async global->LDS pipeline.
__global__ __launch_bounds__(256) void nms_mul_kernel(
    const float* __restrict__ in, const float* __restrict__ probs,
    float* __restrict__ out) {
  const int tid = threadIdx.x;
  const int g   = blockIdx.x * 256 + tid;
  const int p0  = g << 2;
  const int b   = p0 / NHW;
  const int rem = p0 - b * NHW;
  const int h   = rem / NW;
  const int w0  = rem - h * NW;

  const float* pr = probs + (size_t)b * NHW;

  // Neighborhood rows for pixels w0-1 .. w0+4 (zero padded).
  float up[6], mid[6], dn[6];
#pragma unroll
  for (int j = 0; j < 6; ++j) {
    const int w   = w0 - 1 + j;
    const bool wk = (unsigned)w < (unsigned)NW;
    up[j]  = (wk && h > 0)      ? pr[(size_t)(h - 1) * NW + w] : 0.0f;
    mid[j] = wk                 ? pr[(size_t)h * NW + w]       : 0.0f;
    dn[j]  = (wk && h < NH - 1) ? pr[(size_t)(h + 1) * NW + w] : 0.0f;
  }

  // Raster order: neighbors before center use strict >, after center use >=.
  f4 m;
#pragma unroll
  for (int j = 0; j < 4; ++j) {
    const float p = mid[j + 1];
    const bool k = (p >  up[j])  & (p >  up[j + 1]) & (p >  up[j + 2]) &
                   (p >  mid[j]) & (p >= mid[j + 2]) &
                   (p >= dn[j])  & (p >= dn[j + 1]) & (p >= dn[j + 2]);
    m[j] = k ? 1.0f : 0.0f;
  }

  const float* gbase = in  + (size_t)b * NC * NHW + rem;
  float*       obase = out + (size_t)b * NC * NHW + rem;

  // 4-stage double buffer in LDS. Each wave's lanes write/read only their own
  // 16 B chunk -> no workgroup barriers required. Async loads complete in
  // order, so s_wait_asynccnt (NSTAGE-1) guarantees the oldest stage landed.
  __shared__ float smem[NSTAGE][256 * 4];
  const float* lptr[NSTAGE];
  unsigned     lbyte[NSTAGE];
#pragma unroll
  for (int s = 0; s < NSTAGE; ++s) {
    lptr[s]  = &smem[s][tid * 4];
    lbyte[s] = (unsigned)(uintptr_t)lptr[s];  // flat LDS addr low 32b = offset
  }

  // Prologue: fill the pipeline with channels 0..NSTAGE-1.
#pragma unroll
  for (int s = 0; s < NSTAGE; ++s)
    async_g2l_b128(gbase + (size_t)s * NHW, lbyte[s]);

  // Steady state: channels 0 .. NC-NSTAGE-1 (80 iterations, %NSTAGE aligned).
#pragma unroll 4
  for (int c = 0; c < NC - NSTAGE; ++c) {
    wait_async_le3();                        // stage (c % NSTAGE) resident
    f4 v = *(const f4*)lptr[c % NSTAGE];     // ds_load_b128
    wait_ds_0();                             // v in VGPRs; stage reusable
    async_g2l_b128(gbase + (size_t)(c + NSTAGE) * NHW, lbyte[c % NSTAGE]);
    *(f4*)(obase + (size_t)c * NHW) = v * m; // global_store_b128
  }

  // Epilogue: everything issued; drain the last NSTAGE resident stages.
  wait_async_0();
#pragma unroll
  for (int c = NC - NSTAGE; c < NC; ++c) {
    f4 v = *(const f4*)lptr[c % NSTAGE];
    *(f4*)(obase + (size_t)c * NHW) = v * m;
  }
}

extern "C" void kernel_launch(void* const* d_in, const int* in_sizes, int n_in,
                              void* d_out, int out_size, void* d_ws, size_t ws_size,
                              hipStream_t stream) {
  (void)in_sizes; (void)n_in; (void)out_size; (void)ws_size;
  const float* in    = (const float*)d_in[0];
  float*       out   = (float*)d_out;
  float*       probs = (float*)d_ws;     // needs NB*NH*NW*4 = 4 MB of scratch

  const int groups = NB * NHW / 4;       // 262144 four-pixel groups
  dim3 grid(groups / 256), block(256);

  hipLaunchKernelGGL(probs_max_kernel, grid, block, 0, stream, in, probs);
  hipLaunchKernelGGL(nms_mul_kernel,  grid, block, 0, stream, in, probs, out);
}